// Model_25056839205362
// MI455X (gfx1250) — compile-verified
//
#include <hip/hip_runtime.h>
#include <cstddef>

typedef __attribute__((ext_vector_type(2))) float v2f;
typedef __attribute__((ext_vector_type(8))) float v8f;

#define TMAX   20
#define MDIM   512
#define NELEM  (MDIM * MDIM)
#define BLOCK  1024
#define NWAVES (BLOCK / 32)

// One workgroup per batch matrix. Linear-domain Sinkhorn:
//   E = exp(x - max);  repeat: u = 1/(E v), v = 1/(E^T u);  out = u E v
// E is materialized into d_out and scaled in place at the end.
__global__ __launch_bounds__(BLOCK)
void sinkhorn_wmma_kernel(const float* __restrict__ in, float* __restrict__ out)
{
    const int batch = blockIdx.x;
    const float* __restrict__ src = in + (size_t)batch * NELEM;
    float* __restrict__ E = out + (size_t)batch * NELEM;

    __shared__ float uS[MDIM];
    __shared__ float vS[MDIM];
    __shared__ float redS[NWAVES];

    const int tid  = threadIdx.x;
    const int lane = tid & 31;
    const int wave = tid >> 5;

    // ---------- Pass 0a: per-batch global max ----------
    float mx = -3.402823466e+38f;
    const float4* src4 = (const float4*)src;
    for (int k = tid; k < NELEM / 4; k += BLOCK) {
        float4 x = src4[k];
        mx = fmaxf(mx, fmaxf(fmaxf(x.x, x.y), fmaxf(x.z, x.w)));
    }
    #pragma unroll
    for (int off = 16; off > 0; off >>= 1)
        mx = fmaxf(mx, __shfl_xor(mx, off, 32));
    if (lane == 0) redS[wave] = mx;
    __syncthreads();
    if (wave == 0) {
        float m2 = redS[lane];
        #pragma unroll
        for (int off = 16; off > 0; off >>= 1)
            m2 = fmaxf(m2, __shfl_xor(m2, off, 32));
        if (lane == 0) redS[0] = m2;
    }
    __syncthreads();
    mx = redS[0];

    // ---------- Pass 0b: E = exp(x - max), v = 1 ----------
    if (tid < MDIM) vS[tid] = 1.0f;
    float4* E4 = (float4*)E;
    for (int k = tid; k < NELEM / 4; k += BLOCK) {
        float4 x = src4[k];
        float4 e;
        e.x = __expf(x.x - mx);
        e.y = __expf(x.y - mx);
        e.z = __expf(x.z - mx);
        e.w = __expf(x.w - mx);
        E4[k] = e;
    }
    __syncthreads();

    const int half = lane >> 4;   // 0: lanes 0-15, 1: lanes 16-31
    const int jl   = lane & 15;

    for (int t = 0; t < TMAX; ++t) {
        // ---- Row pass (VALU, coalesced float4): u_i = 1 / sum_j E[i][j]*v[j]
        {
            const float4* v4 = (const float4*)vS;
            const int i0 = wave * 16;
            for (int i = i0; i < i0 + 16; ++i) {
                const float4* row4 = (const float4*)(E + (size_t)i * MDIM);
                float p = 0.f;
                #pragma unroll
                for (int c = 0; c < 4; ++c) {
                    float4 a = row4[c * 32 + lane];
                    float4 b = v4[c * 32 + lane];
                    p = fmaf(a.x, b.x, p);
                    p = fmaf(a.y, b.y, p);
                    p = fmaf(a.z, b.z, p);
                    p = fmaf(a.w, b.w, p);
                }
                #pragma unroll
                for (int off = 16; off > 0; off >>= 1)
                    p += __shfl_xor(p, off, 32);
                if (lane == 0) uS[i] = 1.0f / p;
            }
        }
        __syncthreads();

        // ---- Column pass (WMMA f32 16x16x4): v_j = 1 / sum_i E[i][j]*u[i]
        // D[m][n] = sum_k A[m][k]*B[k][n] with A[m][k] = u[kb+k] (broadcast),
        // B[k][n] = E[kb+k][jbase+n]  ->  every row of D = column sums.
        {
            const int jbase = wave * 16;            // 32 waves x 16 cols = 512
            const float* colp = E + jbase + jl;
            v8f acc = {0.f, 0.f, 0.f, 0.f, 0.f, 0.f, 0.f, 0.f};
            for (int kb = 0; kb < MDIM; kb += 4) {
                const int r = kb + 2 * half;        // K rows {0,1} / {2,3}
                v2f a, b;
                a[0] = uS[r];
                a[1] = uS[r + 1];
                b[0] = colp[(size_t)r * MDIM];      // lane-contiguous 64B row segs
                b[1] = colp[(size_t)(r + 1) * MDIM];
                acc = __builtin_amdgcn_wmma_f32_16x16x4_f32(
                        false, a, false, b, (short)0, acc, false, false);
            }
            // D VGPR0, lanes 0-15: M=0 row = the 16 column sums of this wave
            if (half == 0) vS[jbase + jl] = 1.0f / acc[0];
        }
        __syncthreads();
    }

    // ---------- Final pass: out = u_i * E_ij * v_j (in place over E) ----------
    {
        const float4* v4 = (const float4*)vS;
        for (int k = tid; k < NELEM / 4; k += BLOCK) {
            float4 e = E4[k];
            const int i  = (k * 4) >> 9;   // row
            const int j4 = k & 127;        // col/4
            const float ui = uS[i];
            float4 vv = v4[j4];
            float4 o;
            o.x = e.x * ui * vv.x;
            o.y = e.y * ui * vv.y;
            o.z = e.z * ui * vv.z;
            o.w = e.w * ui * vv.w;
            E4[k] = o;
        }
    }
}

extern "C" void kernel_launch(void* const* d_in, const int* in_sizes, int n_in,
                              void* d_out, int out_size, void* d_ws, size_t ws_size,
                              hipStream_t stream)
{
    (void)n_in; (void)out_size; (void)d_ws; (void)ws_size;
    const float* logits = (const float*)d_in[0];
    float* out = (float*)d_out;
    const int nbatch = in_sizes[0] / NELEM;   // 128
    sinkhorn_wmma_kernel<<<dim3(nbatch), dim3(BLOCK), 0, stream>>>(logits, out);
}